// ImageProjection_69982197121559
// MI455X (gfx1250) — compile-verified
//
#include <hip/hip_runtime.h>
#include <hip/hip_bf16.h>

typedef float v2f __attribute__((ext_vector_type(2)));
typedef float v8f __attribute__((ext_vector_type(8)));

#ifndef __has_builtin
#define __has_builtin(x) 0
#endif
#if __has_builtin(__builtin_amdgcn_wmma_f32_16x16x4_f32)
#define USE_WMMA_TRANSFORM 1
#else
#define USE_WMMA_TRANSFORM 0
#endif

// ---------------------------------------------------------------------------
// Prep: per-batch rotation matrix R[d][e] (row-major, 9 floats) and fused
// translation t[e] = -pose . R  (3 floats) -> ws[b*12 .. b*12+11]
// ---------------------------------------------------------------------------
__global__ void prep_rot_kernel(const float* __restrict__ pose,
                                const float* __restrict__ rotq,
                                float* __restrict__ rt, int B) {
    int b = blockIdx.x * blockDim.x + threadIdx.x;
    if (b >= B) return;
    float qx = rotq[b * 4 + 0], qy = rotq[b * 4 + 1];
    float qz = rotq[b * 4 + 2], qw = rotq[b * 4 + 3];
    float inv = 1.0f / sqrtf(qx * qx + qy * qy + qz * qz + qw * qw);
    // normalize + conjugate (q *= [-1,-1,-1,1])
    qx *= -inv; qy *= -inv; qz *= -inv; qw *= inv;
    float R[9];
    R[0] = 1.f - 2.f * qy * qy - 2.f * qz * qz;
    R[1] = 2.f * qx * qy + 2.f * qz * qw;
    R[2] = 2.f * qx * qz - 2.f * qy * qw;
    R[3] = 2.f * qx * qy - 2.f * qz * qw;
    R[4] = 1.f - 2.f * qx * qx - 2.f * qz * qz;
    R[5] = 2.f * qy * qz + 2.f * qx * qw;
    R[6] = 2.f * qx * qz + 2.f * qy * qw;
    R[7] = 2.f * qy * qz - 2.f * qx * qw;
    R[8] = 1.f - 2.f * qx * qx - 2.f * qy * qy;
    float px = pose[b * 3 + 0], py = pose[b * 3 + 1], pz = pose[b * 3 + 2];
#pragma unroll
    for (int e = 0; e < 9; ++e) rt[b * 12 + e] = R[e];
#pragma unroll
    for (int e = 0; e < 3; ++e)
        rt[b * 12 + 9 + e] = -(px * R[0 + e] + py * R[3 + e] + pz * R[6 + e]);
}

// ---------------------------------------------------------------------------
// Main: one thread per point. Transform via V_WMMA_F32_16X16X4_F32 (2 WMMAs
// per wave = 32 points, points laid out as B-matrix columns so lane n's D
// VGPRs 0..2 are point n's x,y,z). Then perspective project + depth-tested
// bilinear gather of 16 channels, skipping out-of-bounds corners.
// ---------------------------------------------------------------------------
__global__ __launch_bounds__(256) void proj_gather_kernel(
    const float* __restrict__ locs,   // [B,N,3]
    const float* __restrict__ image,  // [B,16,512,512]
    const float* __restrict__ depth,  // [B,512,512]
    const float* __restrict__ rt,     // [B,12]
    const float* __restrict__ flp,    // [1]
    float* __restrict__ out,          // [B,N,16]
    int N)
{
    constexpr int C = 16, H = 512, W = 512;
    const size_t HW = (size_t)H * W;
    const int b = blockIdx.y;
    const int lane = threadIdx.x & 31;
    const int p = blockIdx.x * blockDim.x + threadIdx.x;   // point index
    const float fl = flp[0];
    const float* r = rt + b * 12;   // uniform -> scalar loads

    // Per-lane point load (coalesced b96): guarded; garbage lanes masked later.
    float lx = 0.f, ly = 0.f, lz = 0.f;
    if (p < N) {
        const float* lp = locs + ((size_t)b * N + p) * 3;
        lx = lp[0]; ly = lp[1]; lz = lp[2];
    }

    float X, Y, Z;
#if USE_WMMA_TRANSFORM
    // A (16x4, M=output coord, K=input homogeneous coord): A[m][k]=R[k][m],
    // A[m][3]=t[m], rows m>=3 are zero.
    // Layout (ISA 7.12.2): lane m holds A[m][0..1]; lane m+16 holds A[m][2..3].
    const int m = lane & 15;
    const bool hi = lane >= 16;
    v2f A; A[0] = 0.f; A[1] = 0.f;
    if (m < 3) {
        A[0] = hi ? r[6 + m] : r[0 + m];   // R[2][m] : R[0][m]
        A[1] = hi ? r[9 + m] : r[3 + m];   // t[m]    : R[1][m]
    }
    // B (4x16): columns are homogeneous points [x,y,z,1].
    // Layout: VGPR v, lane n -> B[v][n]; lane n+16 -> B[v+2][n].
    const int partner = lane ^ 16;
    const float ox = __shfl(lx, partner, 32);
    const float oy = __shfl(ly, partner, 32);
    const float oz = __shfl(lz, partner, 32);
    v2f B1, B2;
    B1[0] = hi ? oz : lx;       // lanes>=16 need z of point (lane-16)
    B1[1] = hi ? 1.f : ly;
    B2[0] = hi ? lz : ox;       // lanes<16 need x of point (lane+16)
    B2[1] = hi ? 1.f : oy;
    v8f Cz = {};
    v8f D1 = __builtin_amdgcn_wmma_f32_16x16x4_f32(
        false, A, false, B1, (short)0, Cz, false, false);
    v8f D2 = __builtin_amdgcn_wmma_f32_16x16x4_f32(
        false, A, false, B2, (short)0, Cz, false, false);
    // lane<16: own column of D1; lane>=16: column (lane-16) of D2, held by partner.
    const float sx = __shfl(D2[0], partner, 32);
    const float sy = __shfl(D2[1], partner, 32);
    const float sz = __shfl(D2[2], partner, 32);
    X = hi ? sx : D1[0];
    Y = hi ? sy : D1[1];
    Z = hi ? sz : D1[2];
#else
    X = lx * r[0] + ly * r[3] + lz * r[6] + r[9];
    Y = lx * r[1] + ly * r[4] + lz * r[7] + r[10];
    Z = lx * r[2] + ly * r[5] + lz * r[8] + r[11];
#endif

    // Perspective projection (matches reference: zs = z>0 ? z : 1; f{x,y} = p - 0.5)
    const float zs = (Z > 0.f) ? Z : 1.0f;
    const float s = fl / zs;
    float fx = X * s + (0.5f * W - 0.5f);
    float fy = Y * s + (0.5f * H - 0.5f);
    // keep int conversion well-defined for wild projections (cannot affect validity)
    fx = fminf(fmaxf(fx, -65536.f), 65536.f);
    fy = fminf(fmaxf(fy, -65536.f), 65536.f);
    const float x0f = floorf(fx), y0f = floorf(fy);
    const float dx = fx - x0f, dy = fy - y0f;
    const int x0 = (int)x0f, y0 = (int)y0f;

    float acc[C];
#pragma unroll
    for (int c = 0; c < C; ++c) acc[c] = 0.f;

    const float* imb = image + (size_t)b * C * HW;
    const float* dmb = depth + (size_t)b * HW;
    const bool zok = (Z > 0.f) && (p < N);

#pragma unroll
    for (int corner = 0; corner < 4; ++corner) {
        const int cx = corner & 1, cy = corner >> 1;
        const int xi = x0 + cx, yi = y0 + cy;
        const float w = (cx ? dx : 1.f - dx) * (cy ? dy : 1.f - dy);
        if (zok && xi >= 0 && xi < W && yi >= 0 && yi < H) {
            const size_t off = (size_t)yi * W + xi;
            const float dm = dmb[off];
            if (Z <= dm) {
                const float* pp = imb + off;
#pragma unroll
                for (int c = 0; c < C; ++c)
                    acc[c] += w * pp[(size_t)c * HW];
            }
        }
    }

    if (p < N) {
        float4* o4 = (float4*)(out + ((size_t)b * N + p) * C);
        o4[0] = make_float4(acc[0],  acc[1],  acc[2],  acc[3]);
        o4[1] = make_float4(acc[4],  acc[5],  acc[6],  acc[7]);
        o4[2] = make_float4(acc[8],  acc[9],  acc[10], acc[11]);
        o4[3] = make_float4(acc[12], acc[13], acc[14], acc[15]);
    }
}

extern "C" void kernel_launch(void* const* d_in, const int* in_sizes, int n_in,
                              void* d_out, int out_size, void* d_ws, size_t ws_size,
                              hipStream_t stream) {
    const float* locs  = (const float*)d_in[0];
    const float* image = (const float*)d_in[1];
    const float* pose  = (const float*)d_in[2];
    const float* rotq  = (const float*)d_in[3];
    const float* depth = (const float*)d_in[4];
    const float* flp   = (const float*)d_in[5];
    float* out = (float*)d_out;
    float* rt  = (float*)d_ws;                 // B*12 floats of scratch

    const int B = in_sizes[2] / 3;             // camera_pose is [B,3]
    const int N = in_sizes[0] / (3 * B);       // locs is [B,N,3]

    prep_rot_kernel<<<1, 32, 0, stream>>>(pose, rotq, rt, B);

    dim3 grid((N + 255) / 256, B);
    proj_gather_kernel<<<grid, 256, 0, stream>>>(locs, image, depth, rt, flp, out, N);
}